// SingleValueAgent_33681133535598
// MI455X (gfx1250) — compile-verified
//
#include <hip/hip_runtime.h>
#include <cstdint>

// ---------------------------------------------------------------------------
// EMA full-trajectory scan: out[s,t] = a*out[s,t-1] + alpha*x[s,t],
//   alpha = sigmoid(a_raw), a = 1-alpha, x = in[s,t,0]*in[s,t,1]
// Memory-bound: 384 MiB @ 23.3 TB/s ~ 16.5 us floor.
// One wave32 per session; Hillis-Steele weighted scan breaks the serial
// recurrence; gfx1250 async global->LDS ring (ASYNCcnt) provides ~32 MB of
// loads in flight across the grid to cover HBM latency at full bandwidth.
// ---------------------------------------------------------------------------

#define N_SESS   4096
#define N_TRIALS 8192
#define LANES    32
#define WAVES    8                    // waves per block (one session per wave)
#define R        4                    // trials per lane per chunk
#define CHUNK    (LANES * R)          // 128 trials per chunk
#define NCHUNK   (N_TRIALS / CHUNK)   // 64 chunks per session
#define DEPTH    8                    // ring-buffer depth (chunks in flight)
#define CHUNK_BYTES (CHUNK * 2 * 4)   // 1024 B of input per chunk
// steady-state: 2*DEPTH async loads outstanding; oldest chunk done when <= 14
#define ASYNC_WAIT_STEADY "s_wait_asynccnt 14"

typedef float v4f __attribute__((ext_vector_type(4)));

__global__ __launch_bounds__(WAVES * LANES)
void ema_scan_kernel(const float* __restrict__ in,
                     const float* __restrict__ a_raw,
                     float* __restrict__ out)
{
    // LDS ring: per wave, DEPTH chunk buffers of 1 KiB each (64 KiB / block)
    __shared__ v4f buf[WAVES][DEPTH][CHUNK / 2];

    const int lane = threadIdx.x & (LANES - 1);
    const int wv   = threadIdx.x >> 5;
    const int sess = blockIdx.x * WAVES + wv;

    const float ar    = a_raw[0];
    const float alpha = 1.0f / (1.0f + __expf(-ar));
    const float a  = 1.0f - alpha;
    const float a2 = a * a;
    const float a3 = a2 * a;
    const float a4 = a2 * a2;

    // per-lane carry weights: a^{4*lane + i}, i = 1..4  (exact pow-by-squaring)
    float base = 1.0f;
    {
        float p = a4;
        int e = lane;
        while (e) { if (e & 1) base *= p; p *= p; e >>= 1; }
    }
    const float c1 = a  * base;
    const float c2 = a2 * base;
    const float c3 = a3 * base;
    const float c4 = a4 * base;

    const uint64_t gsess =
        (uint64_t)(uintptr_t)in + (uint64_t)sess * (N_TRIALS * 2u * 4u);

    // ---- prologue: fill the async pipeline (2 x B128 per chunk) ----
    #pragma unroll
    for (int d = 0; d < DEPTH; ++d) {
        uint32_t l0 = (uint32_t)(uintptr_t)(&buf[wv][d][0]) + (uint32_t)(lane * 16);
        uint32_t l1 = l0 + (CHUNK_BYTES / 2);
        uint64_t g0 = gsess + (uint64_t)d * CHUNK_BYTES + (uint32_t)(lane * 16);
        uint64_t g1 = g0 + (CHUNK_BYTES / 2);
        asm volatile("global_load_async_to_lds_b128 %0, %1, off"
                     :: "v"(l0), "v"(g0) : "memory");
        asm volatile("global_load_async_to_lds_b128 %0, %1, off"
                     :: "v"(l1), "v"(g1) : "memory");
    }

    float  carry = 0.0f;
    float* osess = out + (size_t)sess * N_TRIALS;

    for (int c = 0; c < NCHUNK; ++c) {
        const int slot = c & (DEPTH - 1);

        // wait until chunk c's two async loads have landed in LDS
        // (async loads complete in order)
        if (c <= NCHUNK - DEPTH) {
            asm volatile(ASYNC_WAIT_STEADY ::: "memory");
        } else {
            asm volatile("s_wait_asynccnt 0" ::: "memory");
        }

        // read this lane's 4 trials (two 16B = 4 x {choice,outcome})
        const v4f* bp = &buf[wv][slot][0];
        v4f u = bp[2 * lane];
        v4f w = bp[2 * lane + 1];

        // make sure our ds reads finished before the slot is overwritten
        asm volatile("s_wait_dscnt 0" ::: "memory");

        // refill this slot with chunk c+DEPTH
        const int nxt = c + DEPTH;
        if (nxt < NCHUNK) {
            uint32_t l0 = (uint32_t)(uintptr_t)(&buf[wv][slot][0]) + (uint32_t)(lane * 16);
            uint64_t g0 = gsess + (uint64_t)nxt * CHUNK_BYTES + (uint32_t)(lane * 16);
            uint32_t l1 = l0 + (CHUNK_BYTES / 2);
            uint64_t g1 = g0 + (CHUNK_BYTES / 2);
            asm volatile("global_load_async_to_lds_b128 %0, %1, off"
                         :: "v"(l0), "v"(g0) : "memory");
            asm volatile("global_load_async_to_lds_b128 %0, %1, off"
                         :: "v"(l1), "v"(g1) : "memory");
        }

        // ---- local (in-lane) recurrence over 4 trials ----
        float x0 = u.x * u.y, x1 = u.z * u.w;
        float x2 = w.x * w.y, x3 = w.z * w.w;
        float y0 = alpha * x0, y1 = alpha * x1, y2 = alpha * x2, y3 = alpha * x3;
        float p0 = y0;
        float p1 = fmaf(a, p0, y1);
        float p2 = fmaf(a, p1, y2);
        float p3 = fmaf(a, p2, y3);

        // ---- wave-level weighted inclusive scan (multiplier a^4) ----
        float B  = p3;
        float pw = a4;
        #pragma unroll
        for (int off = 1; off < LANES; off <<= 1) {
            float t = __shfl_up(B, (unsigned)off, LANES);
            if (lane < off) t = 0.0f;
            B  = fmaf(pw, t, B);
            pw = pw * pw;
        }
        float E = __shfl_up(B, 1u, LANES);     // exclusive: state before this lane
        if (lane == 0) E = 0.0f;

        float s0 = fmaf(a,  E, p0);
        float s1 = fmaf(a2, E, p1);
        float s2 = fmaf(a3, E, p2);
        float s3 = B;

        // fold in the cross-chunk carry
        s0 = fmaf(c1, carry, s0);
        s1 = fmaf(c2, carry, s1);
        s2 = fmaf(c3, carry, s2);
        s3 = fmaf(c4, carry, s3);

        // streaming (non-temporal) coalesced 16B store — write-once output
        v4f o; o.x = s0; o.y = s1; o.z = s2; o.w = s3;
        __builtin_nontemporal_store(
            o, reinterpret_cast<v4f*>(osess + c * CHUNK + 4 * lane));

        carry = __shfl(s3, 31, LANES);         // full state at end of chunk
    }
}

extern "C" void kernel_launch(void* const* d_in, const int* in_sizes, int n_in,
                              void* d_out, int out_size, void* d_ws, size_t ws_size,
                              hipStream_t stream) {
    (void)in_sizes; (void)n_in; (void)out_size; (void)d_ws; (void)ws_size;
    const float* in    = (const float*)d_in[0];   // (4096, 8192, 2) fp32
    const float* a_raw = (const float*)d_in[1];   // (1,) fp32
    float* out = (float*)d_out;                   // (4096, 8192, 1) fp32

    dim3 block(WAVES * LANES);                    // 256 threads = 8 waves
    dim3 grid(N_SESS / WAVES);                    // 512 blocks
    ema_scan_kernel<<<grid, block, 0, stream>>>(in, a_raw, out);
}